// AnimaV4_86887188398446
// MI455X (gfx1250) — compile-verified
//
#include <hip/hip_runtime.h>
#include <math.h>

// ---------------------------------------------------------------------------
// AnimaV4 recurrent agent forward pass for gfx1250 (MI455X).
// One wave32 per block owns 16 batch rows for the full T=256 scan.
// All per-step GEMMs use V_WMMA_F32_16X16X4_F32 (fp32, matches reference).
// Obs tile staged with GLOBAL_LOAD_ASYNC_TO_LDS_B128 when available.
// ---------------------------------------------------------------------------

typedef float v2f __attribute__((ext_vector_type(2)));
typedef float v8f __attribute__((ext_vector_type(8)));
typedef int   vi4 __attribute__((__vector_size__(4 * sizeof(int))));

#define SENSORY   8
#define WORLD     32
#define ACTION    4
#define NSLOTS    4
#define DH        16
#define TSTEPS    256
#define BATCH     4096
#define ROWS      16     // batch rows per wave
#define XS        36     // padded row stride (144B: 16B-aligned rows, conflict-free)
#define GS        100    // padded row stride for 96-wide GRU gate tiles
#define TWO_PI_F  6.28318530717958647692f

#if defined(__has_builtin)
#if __has_builtin(__builtin_amdgcn_global_load_async_to_lds_b128) && \
    __has_builtin(__builtin_amdgcn_s_wait_asynccnt)
#define USE_ASYNC_OBS 1
#endif
#endif

struct KParams {
  const float *obs;
  const float *W_we,*b_we,*W_wp1,*b_wp1,*W_wp2,*b_wp2;
  const float *Wih_w,*bih_w,*Whh_w,*bhh_w;
  const float *W_ee,*b_ee,*Wih_i,*bih_i,*Whh_i,*bhh_i;
  const float *W_ah1,*b_ah1,*W_ah2,*b_ah2;
  const float *W_q,*b_q,*W_k,*b_k,*W_v,*b_v,*W_o,*b_o;
  const float *W_g1,*b_g1,*W_g2,*b_g2,*W_ga,*b_ga;
  const float *blend_w,*W_u1,*b_u1,*W_u2,*b_u2;
  float *out;
};

// Branch-free sigmoid / tanh (v_exp_f32 + v_rcp_f32, no exec-mask branches).
__device__ __forceinline__ float sigm(float x) {
  return __builtin_amdgcn_rcpf(1.0f + __expf(-x));
}
__device__ __forceinline__ float fast_tanh(float x) {
  const float e = __expf(-2.0f * fabsf(x));
  const float r = (1.0f - e) * __builtin_amdgcn_rcpf(1.0f + e);
  return copysignf(r, x);
}

// Y[16 x DOUTP] = act( X[16 x DIN] @ W^T + bias ), W global row-major [dout_real x din_real].
// X,Y in LDS with row strides xs/ys. DOUTP multiple of 16 (>= dout_real, padded cols are junk
// and never consumed). DIN multiple of 4. ACT: 0 = identity, 1 = tanh.
template<int DIN, int DOUTP, int ACT>
__device__ __forceinline__ void gemm16(const float* X, int xs,
                                       const float* __restrict__ Wg,
                                       const float* __restrict__ bg,
                                       int dout_real, int din_real,
                                       float* Y, int ys, int lane)
{
  const int m  = lane & 15;   // A-matrix row / C-matrix column
  const int hi = lane >> 4;   // half-wave selector
#pragma unroll
  for (int nt = 0; nt < DOUTP / 16; ++nt) {
    const int ncol  = nt * 16 + m;
    const int nw    = (ncol < dout_real) ? ncol : (dout_real - 1);   // clamp OOB rows
    const float bia = (ncol < dout_real) ? bg[ncol] : 0.0f;
    v8f c;
#pragma unroll
    for (int i = 0; i < 8; ++i) c[i] = bia;
#pragma unroll
    for (int k0 = 0; k0 < DIN; k0 += 4) {
      const int ka = k0 + 2 * hi;                 // ISA A/B fragment layout (16x4 f32)
      v2f a, b;
      a.x = X[m * xs + ka];
      a.y = X[m * xs + ka + 1];
      b.x = Wg[nw * din_real + ka];               // B(k,n) = W(n,k)
      b.y = Wg[nw * din_real + ka + 1];
      c = __builtin_amdgcn_wmma_f32_16x16x4_f32(false, a, false, b,
                                                (short)0, c, false, false);
    }
#pragma unroll
    for (int i = 0; i < 8; ++i) {                 // C/D: row = hi*8+i, col = ncol
      float v = c[i];
      if (ACT == 1) v = fast_tanh(v);
      Y[(hi * 8 + i) * ys + ncol] = v;
    }
  }
}

__device__ __forceinline__ void gru_update(float* S, const float* GI, const float* GH, int lane)
{
  const int m = lane & 15, hi = lane >> 4;
#pragma unroll
  for (int i = 0; i < 16; ++i) {
    const int c = hi * 16 + i;
    const float ir = GI[m * GS + c],       hr = GH[m * GS + c];
    const float iz = GI[m * GS + 32 + c],  hz = GH[m * GS + 32 + c];
    const float in = GI[m * GS + 64 + c],  hn = GH[m * GS + 64 + c];
    const float r = sigm(ir + hr);
    const float z = sigm(iz + hz);
    const float n = fast_tanh(in + r * hn);
    const float old = S[m * XS + c];
    S[m * XS + c] = (1.0f - z) * n + z * old;
  }
}

__global__ __launch_bounds__(32)
void anima_v4_kernel(KParams p)
{
  __shared__ float sW[ROWS * XS];                // world state w
  __shared__ float sH[ROWS * XS];                // internal state h
  __shared__ float sMEM[NSLOTS * ROWS * XS];     // memory slots
  __shared__ float sOBS[ROWS * XS];
  __shared__ float sA[ROWS * XS], sB[ROWS * XS], sC[ROWS * XS], sD[ROWS * XS];
  __shared__ float sGI[ROWS * GS], sGH[ROWS * GS];
  __shared__ float sPh[ROWS], sEm[ROWS], sAct[ROWS * ACTION];
  __shared__ int   sCy[ROWS], sWr[ROWS], sSlot[ROWS];

  const int lane    = threadIdx.x;
  const int m       = lane & 15;
  const int hi      = lane >> 4;
  const int rowBase = blockIdx.x * ROWS;

  for (int i = lane; i < ROWS * XS; i += 32) { sW[i] = 0.0f; sH[i] = 0.0f; }
  for (int i = lane; i < NSLOTS * ROWS * XS; i += 32) sMEM[i] = 0.0f;
  if (lane < ROWS) { sPh[lane] = 0.0f; sCy[lane] = 0; }
  __syncthreads();

  const float blend = sigm(p.blend_w[0]);
  // Hoist loop-invariant urgency parameters (uniform scalar loads).
  float u1w[8], u1b[8], u2w[8];
#pragma unroll
  for (int j = 0; j < 8; ++j) {
    u1w[j] = p.W_u1[j]; u1b[j] = p.b_u1[j]; u2w[j] = p.W_u2[j];
  }
  const float u2b = p.b_u2[0];

#pragma unroll 1
  for (int t = 0; t < TSTEPS; ++t) {
    // ---- stage obs tile: 16 rows x 8 floats, contiguous 512B -> LDS ------
    {
      const float* op = p.obs + ((size_t)t * BATCH + rowBase) * SENSORY;
#ifdef USE_ASYNC_OBS
      // Each lane moves 16B: lane even -> cols 0..3, lane odd -> cols 4..7 of row lane>>1.
      // LDS row stride = 144B, so every destination is 16B-aligned.
      const int row  = lane >> 1;
      const int col0 = (lane & 1) * 4;
      __builtin_amdgcn_global_load_async_to_lds_b128(
          (vi4*)(op + lane * 4),
          (vi4*)&sOBS[row * XS + col0],
          0, 0);
      __builtin_amdgcn_s_wait_asynccnt(0);
#else
      const float4 f4 = ((const float4*)op)[lane];
      int g = lane * 4;
      sOBS[(g >> 3) * XS + (g & 7)] = f4.x; ++g;
      sOBS[(g >> 3) * XS + (g & 7)] = f4.y; ++g;
      sOBS[(g >> 3) * XS + (g & 7)] = f4.z; ++g;
      sOBS[(g >> 3) * XS + (g & 7)] = f4.w;
#endif
      if (t + 1 < TSTEPS)
        __builtin_prefetch(p.obs + ((size_t)(t + 1) * BATCH + rowBase) * SENSORY + lane * 4, 0, 0);
    }
    __syncthreads();

    // ---- obs_enc = tanh(obs @ W_we^T + b_we) -> sA -----------------------
    gemm16<8, 32, 1>(sOBS, XS, p.W_we, p.b_we, 32, 8, sA, XS, lane);
    __syncthreads();

    // ---- MHA over memory slots ------------------------------------------
    gemm16<32, 32, 0>(sW, XS, p.W_q, p.b_q, 32, 32, sB, XS, lane);   // q -> sB
    __syncthreads();

    float sc[NSLOTS];
#pragma unroll
    for (int s = 0; s < NSLOTS; ++s) {
      gemm16<32, 32, 0>(&sMEM[s * ROWS * XS], XS, p.W_k, p.b_k, 32, 32, sC, XS, lane);
      __syncthreads();
      float acc = 0.0f;
#pragma unroll
      for (int d = 0; d < DH; ++d)
        acc += sB[m * XS + hi * DH + d] * sC[m * XS + hi * DH + d];
      sc[s] = acc * 0.25f;  // 1/sqrt(16)
      __syncthreads();
    }
    {
      const float mx = fmaxf(fmaxf(sc[0], sc[1]), fmaxf(sc[2], sc[3]));
      float e[NSLOTS]; float sum = 0.0f;
#pragma unroll
      for (int s = 0; s < NSLOTS; ++s) { e[s] = __expf(sc[s] - mx); sum += e[s]; }
      const float inv = __builtin_amdgcn_rcpf(sum);
#pragma unroll
      for (int s = 0; s < NSLOTS; ++s) sc[s] = e[s] * inv;  // attn weights
    }
    float oacc[DH];
#pragma unroll
    for (int d = 0; d < DH; ++d) oacc[d] = 0.0f;
#pragma unroll
    for (int s = 0; s < NSLOTS; ++s) {
      gemm16<32, 32, 0>(&sMEM[s * ROWS * XS], XS, p.W_v, p.b_v, 32, 32, sC, XS, lane);
      __syncthreads();
#pragma unroll
      for (int d = 0; d < DH; ++d)
        oacc[d] += sc[s] * sC[m * XS + hi * DH + d];
      __syncthreads();
    }
#pragma unroll
    for (int d = 0; d < DH; ++d) sD[m * XS + hi * DH + d] = oacc[d];
    __syncthreads();
    gemm16<32, 32, 0>(sD, XS, p.W_o, p.b_o, 32, 32, sC, XS, lane);   // context -> sC
    __syncthreads();

    // ---- x = obs_enc + 0.1*context -> sD --------------------------------
#pragma unroll
    for (int i = 0; i < 16; ++i) {
      const int c = hi * 16 + i;
      sD[m * XS + c] = sA[m * XS + c] + 0.1f * sC[m * XS + c];
    }
    __syncthreads();

    // ---- world GRU -------------------------------------------------------
    gemm16<32, 96, 0>(sD, XS, p.Wih_w, p.bih_w, 96, 32, sGI, GS, lane);
    gemm16<32, 96, 0>(sW, XS, p.Whh_w, p.bhh_w, 96, 32, sGH, GS, lane);
    __syncthreads();
    gru_update(sW, sGI, sGH, lane);
    __syncthreads();

    // ---- g path (uses obs[:, :4], before obs buffer is reused) ----------
    gemm16<4, 16, 1>(sOBS, XS, p.W_g1, p.b_g1, 16, 4, sA, XS, lane);
    __syncthreads();
    gemm16<16, 16, 1>(sA, XS, p.W_g2, p.b_g2, 16, 16, sB, XS, lane);
    __syncthreads();
    gemm16<16, 16, 1>(sB, XS, p.W_ga, p.b_ga, 4, 16, sC, XS, lane);  // g_act cols 0..3 -> sC
    __syncthreads();

    // ---- prediction + error ---------------------------------------------
    gemm16<32, 32, 1>(sW, XS, p.W_wp1, p.b_wp1, 32, 32, sA, XS, lane);
    __syncthreads();
    gemm16<32, 16, 0>(sA, XS, p.W_wp2, p.b_wp2, 8, 32, sB, XS, lane); // pred cols 0..7 -> sB
    __syncthreads();
    if (lane < ROWS) {
      float s = 0.0f;
#pragma unroll
      for (int c2 = 0; c2 < SENSORY; ++c2) {
        const float e = sOBS[lane * XS + c2] - sB[lane * XS + c2];
        sD[lane * XS + c2] = e;
        s += fabsf(e);
      }
      sEm[lane] = s * 0.125f;
    }
    __syncthreads();

    // ---- internal GRU ----------------------------------------------------
    gemm16<8, 32, 1>(sD, XS, p.W_ee, p.b_ee, 32, 8, sA, XS, lane);   // err_enc -> sA
    __syncthreads();
    gemm16<32, 96, 0>(sA, XS, p.Wih_i, p.bih_i, 96, 32, sGI, GS, lane);
    gemm16<32, 96, 0>(sH, XS, p.Whh_i, p.bhh_i, 96, 32, sGH, GS, lane);
    __syncthreads();
    gru_update(sH, sGI, sGH, lane);
    __syncthreads();

    // ---- base action -----------------------------------------------------
    gemm16<32, 32, 1>(sH, XS, p.W_ah1, p.b_ah1, 32, 32, sA, XS, lane);
    __syncthreads();
    gemm16<32, 16, 1>(sA, XS, p.W_ah2, p.b_ah2, 4, 32, sB, XS, lane); // base cols 0..3 -> sB
    __syncthreads();

    // ---- blend, urgency, phase/cycle (one lane per row) -----------------
    if (lane < ROWS) {
      const float em = sEm[lane];
      float u = u2b;
#pragma unroll
      for (int j = 0; j < 8; ++j)
        u += fast_tanh(em * u1w[j] + u1b[j]) * u2w[j];
      const float scale = 1.0f + sigm(u);
      float activity = 0.0f;
#pragma unroll
      for (int c2 = 0; c2 < ACTION; ++c2) {
        float a = (1.0f - blend) * sB[lane * XS + c2] + blend * sC[lane * XS + c2];
        if (em > 0.5f) a *= scale;
        sAct[lane * ACTION + c2] = a;
        activity += fabsf(a);
      }
      activity *= 0.25f;
      float ph = sPh[lane] + 0.1f * (1.0f + 0.2f * activity);
      const int wrap = (ph >= TWO_PI_F) ? 1 : 0;
      if (wrap) ph -= TWO_PI_F;
      const int cy = sCy[lane] + wrap;
      sPh[lane] = ph; sCy[lane] = cy;
      sWr[lane] = wrap; sSlot[lane] = cy & 3;
    }
    __syncthreads();

    // ---- memory slot write (all lanes) ----------------------------------
    {
      const int wrap = sWr[m];
      const int slot = sSlot[m];
      if (wrap) {
#pragma unroll
        for (int i = 0; i < 16; ++i) {
          const int c = hi * 16 + i;
          sMEM[slot * ROWS * XS + m * XS + c] = sW[m * XS + c];
        }
      }
    }

    // ---- store actions: 16 rows x 4 = 64 floats, 2 per lane -------------
    {
      const int row = lane >> 1;
      const int c0  = (lane & 1) * 2;
      float* o = p.out + ((size_t)t * BATCH + rowBase + row) * ACTION + c0;
      o[0] = sAct[row * ACTION + c0];
      o[1] = sAct[row * ACTION + c0 + 1];
    }
    __syncthreads();
  }
}

extern "C" void kernel_launch(void* const* d_in, const int* in_sizes, int n_in,
                              void* d_out, int out_size, void* d_ws, size_t ws_size,
                              hipStream_t stream) {
  (void)in_sizes; (void)n_in; (void)out_size; (void)d_ws; (void)ws_size;
  KParams kp;
  kp.obs   = (const float*)d_in[0];
  kp.W_we  = (const float*)d_in[1];  kp.b_we  = (const float*)d_in[2];
  kp.W_wp1 = (const float*)d_in[3];  kp.b_wp1 = (const float*)d_in[4];
  kp.W_wp2 = (const float*)d_in[5];  kp.b_wp2 = (const float*)d_in[6];
  kp.Wih_w = (const float*)d_in[7];  kp.bih_w = (const float*)d_in[8];
  kp.Whh_w = (const float*)d_in[9];  kp.bhh_w = (const float*)d_in[10];
  kp.W_ee  = (const float*)d_in[11]; kp.b_ee  = (const float*)d_in[12];
  kp.Wih_i = (const float*)d_in[13]; kp.bih_i = (const float*)d_in[14];
  kp.Whh_i = (const float*)d_in[15]; kp.bhh_i = (const float*)d_in[16];
  kp.W_ah1 = (const float*)d_in[17]; kp.b_ah1 = (const float*)d_in[18];
  kp.W_ah2 = (const float*)d_in[19]; kp.b_ah2 = (const float*)d_in[20];
  kp.W_q   = (const float*)d_in[21]; kp.b_q   = (const float*)d_in[22];
  kp.W_k   = (const float*)d_in[23]; kp.b_k   = (const float*)d_in[24];
  kp.W_v   = (const float*)d_in[25]; kp.b_v   = (const float*)d_in[26];
  kp.W_o   = (const float*)d_in[27]; kp.b_o   = (const float*)d_in[28];
  kp.W_g1  = (const float*)d_in[29]; kp.b_g1  = (const float*)d_in[30];
  kp.W_g2  = (const float*)d_in[31]; kp.b_g2  = (const float*)d_in[32];
  kp.W_ga  = (const float*)d_in[33]; kp.b_ga  = (const float*)d_in[34];
  kp.blend_w = (const float*)d_in[35];
  kp.W_u1  = (const float*)d_in[36]; kp.b_u1  = (const float*)d_in[37];
  kp.W_u2  = (const float*)d_in[38]; kp.b_u2  = (const float*)d_in[39];
  kp.out   = (float*)d_out;

  anima_v4_kernel<<<dim3(BATCH / ROWS), dim3(32), 0, stream>>>(kp);
}